// Basket_Embedding_61641370632879
// MI455X (gfx1250) — compile-verified
//
#include <hip/hip_runtime.h>

// Basket mean-pooling (B=1024, L=50, M=20, VOCAB=100000, H=64).
// Gather-bound: ~131MB of L2-resident row gathers. Staging via CDNA5
// GLOBAL_LOAD_ASYNC_TO_LDS_B128 (ASYNCcnt), masked reduction via
// V_WMMA_F32_16X16X4_F32.

typedef __attribute__((ext_vector_type(2))) float v2f;
typedef __attribute__((ext_vector_type(8))) float v8f;
typedef __attribute__((address_space(3))) char  lds_char;

#define NB         (1024 * 50)  // total baskets
#define M_ITEMS    20
#define H_DIM      64
#define ROW_STRIDE 80           // LDS row stride in floats (320B; 256B data + 64B pad)

__global__ __launch_bounds__(256) void basket_pool_wmma(
    const int*   __restrict__ item_ids,     // [NB, 20]
    const int*   __restrict__ basket_lens,  // [NB]
    const float* __restrict__ emb,          // [VOCAB, 64]
    float*       __restrict__ out)          // [NB, 64]
{
    __shared__ __align__(16) float tile[8 * M_ITEMS * ROW_STRIDE]; // 50 KB

    const int lane = threadIdx.x & 31;
    const int wave = threadIdx.x >> 5;
    const int half = lane >> 4;     // 0: lanes 0-15 (row m / K0,K1), 1: lanes 16-31 (row m+1 / K2,K3)
    const int col  = lane & 15;

    const int bl = blockIdx.x * 8 + wave;   // one basket per wave (grid covers NB exactly)
    if (bl >= NB) return;                    // wave-uniform

    const int len = basket_lens[bl];
    const int* __restrict__ ids = item_ids + bl * M_ITEMS;
    float* __restrict__ t = tile + wave * (M_ITEMS * ROW_STRIDE);

    // LDS byte address of this wave's tile (raw addrspace(3) offset for the async op).
    const unsigned lds_base = (unsigned)(unsigned long long)(lds_char*)t;
    const unsigned long long emb_base = (unsigned long long)emb;

    // All 20 ids up front: 5 uniform int4 loads (80B, 16B-aligned).
    int idv[M_ITEMS];
    {
        const int4* __restrict__ ids4 = (const int4*)ids;
        #pragma unroll
        for (int q = 0; q < 5; ++q) {
            const int4 w = ids4[q];
            idv[4*q + 0] = w.x; idv[4*q + 1] = w.y;
            idv[4*q + 2] = w.z; idv[4*q + 3] = w.w;
        }
    }

    // ---- Stage rows into LDS: one async B128 moves TWO rows (512B) straight
    // global->LDS, tracked by ASYNCcnt. Pairs entirely past `len` are zero-filled
    // (poison/stale LDS could be NaN; 0*NaN would poison the WMMA). A pair whose
    // second row is just past `len` loads a real (finite) row that the A-mask kills.
    #pragma unroll
    for (int m = 0; m < M_ITEMS; m += 2) {
        if (m < len) {  // wave-uniform
            const unsigned goff  = (unsigned)(half ? idv[m + 1] : idv[m]) * (H_DIM * 4u)
                                 + (unsigned)col * 16u;
            const unsigned laddr = lds_base + (unsigned)(m + half) * (ROW_STRIDE * 4u)
                                 + (unsigned)col * 16u;
            asm volatile("global_load_async_to_lds_b128 %0, %1, %2"
                         :: "v"(laddr), "v"(goff), "s"(emb_base)
                         : "memory");
        } else {
            const float4 z = make_float4(0.f, 0.f, 0.f, 0.f);
            *((float4*)(t + (m + half) * ROW_STRIDE) + col) = z;   // ds_store_b128
        }
    }

    // Fence: all async global->LDS transfers complete before LDS reads below.
    asm volatile("s_wait_asynccnt 0x0" ::: "memory");

    const float inv = 1.0f / (float)(len > 0 ? len : 1);

    // ---- Masked reduction via WMMA f32 16x16x4.
    // A (16x4): only matrix row 0 nonzero -> lane 0 holds mask[K0,K1], lane 16 mask[K2,K3].
    // B (4x16): lanes 0-15 = N; VGPR0 = K0 (half 0) / K2 (half 1), VGPR1 = K1/K3.
    // D row 0 lands in c[0] on lanes 0-15 (N = col).
    for (int j = 0; j < 4; ++j) {           // 4 chunks of 16 h-values
        v8f c = {};
        const float* tj = t + j * 16 + col;
        #pragma unroll
        for (int s = 0; s < 5; ++s) {       // 5 K-steps of 4 cover M=20 exactly
            const int m0 = 4 * s + 2 * half;
            v2f a;
            a.x = (col == 0 && m0     < len) ? 1.0f : 0.0f;
            a.y = (col == 0 && m0 + 1 < len) ? 1.0f : 0.0f;
            v2f b;
            b.x = tj[(m0    ) * ROW_STRIDE];
            b.y = tj[(m0 + 1) * ROW_STRIDE];
            c = __builtin_amdgcn_wmma_f32_16x16x4_f32(
                    /*neg_a=*/false, a, /*neg_b=*/false, b,
                    /*c_mod=*/(short)0, c, /*reuse_a=*/false, /*reuse_b=*/false);
        }
        if (half == 0) {                    // reconverges before next j's WMMA
            out[(size_t)bl * H_DIM + j * 16 + col] = c[0] * inv;
        }
    }
}

extern "C" void kernel_launch(void* const* d_in, const int* in_sizes, int n_in,
                              void* d_out, int out_size, void* d_ws, size_t ws_size,
                              hipStream_t stream) {
    (void)in_sizes; (void)n_in; (void)out_size; (void)d_ws; (void)ws_size;
    const int*   item_ids    = (const int*)d_in[0];
    const int*   basket_lens = (const int*)d_in[1];
    const float* emb         = (const float*)d_in[2];
    float*       out         = (float*)d_out;

    dim3 grid((NB + 7) / 8);   // 6400 blocks * 8 waves = 51200 baskets, one per wave
    dim3 block(256);
    hipLaunchKernelGGL(basket_pool_wmma, grid, block, 0, stream,
                       item_ids, basket_lens, emb, out);
}